// YOLOLoss_22960895164530
// MI455X (gfx1250) — compile-verified
//
#include <hip/hip_runtime.h>
#include <hip/hip_bf16.h>

#define B_BOXES 2
#define NUM_CLASSES 20
#define PRED_CH 30   // B_BOXES*5 + NUM_CLASSES
#define TGT_CH 25    // 5 + NUM_CLASSES
#define LAMBDA_COORD 5.0f
#define LAMBDA_NOOBJ 0.5f

typedef float v2f __attribute__((ext_vector_type(2)));
typedef float v8f __attribute__((ext_vector_type(8)));

// Wave32 all-reduce using the CDNA5 matrix pipe:
//   A = ones(16x4), B slot (k,n) holds each lane's partial exactly once
//   (B vgpr0 = partial, B vgpr1 = 0) => D[m][n] = column-sum of B.
//   Sum of D row 0 over n = sum of ALL B slots = wave total, independent of
//   the exact B lane->(k,n) mapping. D vgpr0 lane L = colsum[L % 16], and the
//   two 16-lane halves are identical, so 4 xor-shuffles finish the reduce.
__device__ __forceinline__ float wave_reduce_wmma(float acc) {
    v2f a; a[0] = 1.0f; a[1] = 1.0f;     // A = all ones (16x4 f32, 2 VGPRs)
    v2f b; b[0] = acc;  b[1] = 0.0f;     // each lane contributes one B slot
    v8f c = {};
    v8f d = __builtin_amdgcn_wmma_f32_16x16x4_f32(
        /*neg_a=*/false, a, /*neg_b=*/false, b,
        /*c_mod=*/(short)0, c, /*reuse_a=*/false, /*reuse_b=*/false);
    float r = d[0];                      // lane L: colsum[L & 15]
    r += __shfl_xor(r, 1);
    r += __shfl_xor(r, 2);
    r += __shfl_xor(r, 4);
    r += __shfl_xor(r, 8);               // xor-group {0..15}/{16..31}: full total
    return r;
}

__global__ __launch_bounds__(256)
void yolo_loss_partial(const float* __restrict__ preds,
                       const float* __restrict__ targets,
                       float* __restrict__ block_partials,
                       int n_cells) {
    const int cell = blockIdx.x * 256 + threadIdx.x;

    float acc = 0.0f;
    if (cell < n_cells) {
        // preds: 30 f32/cell, cell*120B -> 8-byte aligned: 15x b64 loads
        const float2* p2 = reinterpret_cast<const float2*>(preds + (size_t)cell * PRED_CH);
        float pr[PRED_CH];
#pragma unroll
        for (int i = 0; i < PRED_CH / 2; ++i) {
            float2 v = p2[i];
            pr[2 * i]     = v.x;
            pr[2 * i + 1] = v.y;
        }
        // targets: 25 f32/cell, cell*100B -> only 4-byte aligned: b32 loads
        const float* t = targets + (size_t)cell * TGT_CH;
        float tg[TGT_CH];
#pragma unroll
        for (int i = 0; i < TGT_CH; ++i) tg[i] = t[i];

        const float s0 = pr[0];          // box0 confidence
        const float s5 = pr[5];          // box1 confidence
        // jnp.argmax returns first index on ties -> box1 only if strictly greater
        const bool use1 = (s5 > s0);

        float box_loss = 0.0f;
#pragma unroll
        for (int k = 1; k < 5; ++k) {
            const float bk = use1 ? pr[5 + k] : pr[k];
            const float dk = bk - tg[k];
            box_loss += dk * dk;
        }
        const float b0 = use1 ? s5 : s0;
        const float dpc = b0 - tg[0];
        const float pc_loss = dpc * dpc;

        float class_loss = 0.0f;
#pragma unroll
        for (int j = 0; j < NUM_CLASSES; ++j) {
            const float dj = pr[B_BOXES * 5 + j] - tg[5 + j];
            class_loss += dj * dj;
        }

        const float obj_term   = LAMBDA_COORD * box_loss + pc_loss + class_loss;
        const float noobj_term = LAMBDA_NOOBJ * (s0 * s0 + s5 * s5);

        acc = (tg[0] == 1.0f) ? obj_term : noobj_term;
    }
    // Re-converged here: EXEC is all-1s for the WMMA (required).
    const float wave_sum = wave_reduce_wmma(acc);

    __shared__ float red[8];             // 256 threads = 8 wave32
    const int lane = threadIdx.x & 31;
    const int wid  = threadIdx.x >> 5;
    if (lane == 0) red[wid] = wave_sum;
    __syncthreads();
    if (threadIdx.x == 0) {
        float s = 0.0f;
#pragma unroll
        for (int w = 0; w < 8; ++w) s += red[w];   // fixed order: deterministic
        block_partials[blockIdx.x] = s;
    }
}

__global__ __launch_bounds__(1024)
void yolo_loss_final(const float* __restrict__ partials, int n, float* __restrict__ out) {
    __shared__ float sh[1024];
    float s = 0.0f;
    for (int i = threadIdx.x; i < n; i += 1024) s += partials[i];  // fixed order
    sh[threadIdx.x] = s;
    __syncthreads();
#pragma unroll
    for (int off = 512; off > 0; off >>= 1) {
        if (threadIdx.x < off) sh[threadIdx.x] += sh[threadIdx.x + off];
        __syncthreads();
    }
    if (threadIdx.x == 0) out[0] = sh[0];
}

extern "C" void kernel_launch(void* const* d_in, const int* in_sizes, int n_in,
                              void* d_out, int out_size, void* d_ws, size_t ws_size,
                              hipStream_t stream) {
    const float* preds   = (const float*)d_in[0];
    const float* targets = (const float*)d_in[1];
    float* out      = (float*)d_out;
    float* partials = (float*)d_ws;

    const int n_cells = in_sizes[0] / PRED_CH;          // 16384*49 = 802816
    const int blocks  = (n_cells + 255) / 256;          // 3136 (exact division)

    yolo_loss_partial<<<blocks, 256, 0, stream>>>(preds, targets, partials, n_cells);
    yolo_loss_final<<<1, 1024, 0, stream>>>(partials, blocks, out);
}